// AromaticOxidationNetwork_90031104459452
// MI455X (gfx1250) — compile-verified
//
#include <hip/hip_runtime.h>
#include <hip/hip_bf16.h>

typedef __attribute__((ext_vector_type(2))) float v2f;
typedef __attribute__((ext_vector_type(8))) float v8f;
typedef int v4i_vs __attribute__((vector_size(16)));   // matches builtin param type

#define NROWS 4096   // B*N
#define H 64
#define FD 32
#define NNODE 512

#if defined(__has_builtin)
# if __has_builtin(__builtin_amdgcn_global_load_async_to_lds_b128) && \
     __has_builtin(__builtin_amdgcn_s_wait_asynccnt)
#  define HAVE_ASYNC_LDS 1
# endif
#endif
#ifndef HAVE_ASYNC_LDS
# define HAVE_ASYNC_LDS 0
#endif

__device__ __forceinline__ float silu_f(float x) {
    return x / (1.0f + __expf(-x));
}

// ---------------------------------------------------------------------------
// Encoder: h = silu(LayerNorm((features*fw) @ w_enc + b_enc))
// One wave per 16-row tile; 4 WMMA N-tiles of 16; K=32 in 8 steps of 4.
// ---------------------------------------------------------------------------
__global__ __launch_bounds__(32) void enc_kernel(
    const float* __restrict__ feat, const float* __restrict__ fw,
    const float* __restrict__ w_enc, const float* __restrict__ b_enc,
    const float* __restrict__ ln_g, const float* __restrict__ ln_b,
    float* __restrict__ h)
{
    const int lane = threadIdx.x;
    const int half = lane >> 4, lm = lane & 15;
    const int rowbase = blockIdx.x * 16;

    v8f c[4] = {};   // four 16-wide N tiles of the 16x64 output stripe

    for (int k0 = 0; k0 < FD; k0 += 4) {
        const int ka = k0 + 2 * half;           // this lane-half's K pair
        const float* frow = feat + (size_t)(rowbase + lm) * FD;
        v2f a;
        a.x = frow[ka]     * fw[ka];
        a.y = frow[ka + 1] * fw[ka + 1];
#pragma unroll
        for (int t = 0; t < 4; ++t) {
            v2f bf;
            bf.x = w_enc[(size_t)ka * H       + t * 16 + lm];
            bf.y = w_enc[(size_t)(ka + 1) * H + t * 16 + lm];
            c[t] = __builtin_amdgcn_wmma_f32_16x16x4_f32(
                false, a, false, bf, (short)0, c[t], false, false);
        }
    }

    // + b_enc (per output column)
#pragma unroll
    for (int t = 0; t < 4; ++t) {
        float bias = b_enc[t * 16 + lm];
#pragma unroll
        for (int r = 0; r < 8; ++r) c[t][r] += bias;
    }

    // LayerNorm over H per row (row M = r + 8*half, values across 16 lanes x 4 tiles)
#pragma unroll
    for (int r = 0; r < 8; ++r) {
        float s = 0.f, sq = 0.f;
#pragma unroll
        for (int t = 0; t < 4; ++t) { float v = c[t][r]; s += v; sq += v * v; }
#pragma unroll
        for (int m = 1; m < 16; m <<= 1) {    // masks 1,2,4,8: stay within 16-lane half
            s  += __shfl_xor(s,  m, 32);
            sq += __shfl_xor(sq, m, 32);
        }
        const float mean = s * (1.0f / 64.0f);
        const float var  = sq * (1.0f / 64.0f) - mean * mean;
        const float rs   = rsqrtf(var + 1e-5f);
        const int   row  = rowbase + r + 8 * half;
#pragma unroll
        for (int t = 0; t < 4; ++t) {
            const int n = t * 16 + lm;
            float v = (c[t][r] - mean) * rs * ln_g[n] + ln_b[n];
            h[(size_t)row * H + n] = silu_f(v);
        }
    }
}

// ---------------------------------------------------------------------------
// out[4096x64] = A[4096x64] @ W[64x64]  (+ optional per-column bias)
// ---------------------------------------------------------------------------
__global__ __launch_bounds__(32) void gemm64_kernel(
    const float* __restrict__ A, const float* __restrict__ W,
    const float* __restrict__ bias, int has_bias, float* __restrict__ out)
{
    const int lane = threadIdx.x;
    const int half = lane >> 4, lm = lane & 15;
    const int rowbase = blockIdx.x * 16;

    v8f c[4] = {};
    for (int k0 = 0; k0 < H; k0 += 4) {
        const int ka = k0 + 2 * half;
        const float* arow = A + (size_t)(rowbase + lm) * H;
        v2f a;
        a.x = arow[ka];
        a.y = arow[ka + 1];
#pragma unroll
        for (int t = 0; t < 4; ++t) {
            v2f bf;
            bf.x = W[(size_t)ka * H       + t * 16 + lm];
            bf.y = W[(size_t)(ka + 1) * H + t * 16 + lm];
            c[t] = __builtin_amdgcn_wmma_f32_16x16x4_f32(
                false, a, false, bf, (short)0, c[t], false, false);
        }
    }
#pragma unroll
    for (int t = 0; t < 4; ++t) {
        const float bv = has_bias ? bias[t * 16 + lm] : 0.0f;
#pragma unroll
        for (int r = 0; r < 8; ++r)
            out[(size_t)(rowbase + r + 8 * half) * H + t * 16 + lm] = c[t][r] + bv;
    }
}

// ---------------------------------------------------------------------------
// deg[row] = sum_j adj[row][j] + eps     (one wave per row)
// ---------------------------------------------------------------------------
__global__ __launch_bounds__(256) void deg_kernel(const float* __restrict__ adj,
                                                  float* __restrict__ deg)
{
    const int wave = threadIdx.x >> 5;
    const int lane = threadIdx.x & 31;
    const int row  = blockIdx.x * 8 + wave;
    const float* arow = adj + (size_t)row * NNODE;
    float s = 0.f;
    for (int j = lane; j < NNODE; j += 32) s += arow[j];
#pragma unroll
    for (int m = 1; m < 32; m <<= 1) s += __shfl_xor(s, m, 32);
    if (lane == 0) deg[row] = s + 1e-8f;
}

// ---------------------------------------------------------------------------
// Message pass: h[row,hd] += 0.5/deg * sum_j adj[row,j] * silu(a[row,hd] + b[j,hd])
// (msg bias is pre-folded into b by gemm64_kernel.)
// Block: 256 threads = 4 rows x 64 h-dims.
// b tiles stream memory->LDS via async b128 loads, double-buffered (33 KB LDS).
// ---------------------------------------------------------------------------
#define MSG_ROWS 4
__global__ __launch_bounds__(256) void msg_kernel(
    float* __restrict__ h, const float* __restrict__ abuf,
    const float* __restrict__ bbuf, const float* __restrict__ adj,
    const float* __restrict__ deg)
{
    __shared__ float bs_s[2][64][H];          // 2 x 16 KB, 256B rows (b128-aligned)
    __shared__ float adj_s[MSG_ROWS][64];

    const int tid = threadIdx.x;
    const int r   = tid >> 6;
    const int hd  = tid & 63;
    const int row   = blockIdx.x * MSG_ROWS + r;
    const int batch = blockIdx.x >> 7;        // 128 blocks per batch item

    const float  a_val = abuf[(size_t)row * H + hd];
    const float* bbase = bbuf + (size_t)batch * NNODE * H;
    float acc = 0.0f;

    auto do_chunk = [&](const float (*bt)[H]) {
#pragma unroll 4
        for (int j = 0; j < 64; ++j) {
            const float w = adj_s[r][j];      // wave-uniform per row-group
            if (w != 0.0f) {
                acc += w * silu_f(a_val + bt[j][hd]);
            }
        }
    };

#if HAVE_ASYNC_LDS
    // Each thread owns 4 contiguous-16B slots per 64x64 chunk (1024 b128 / chunk).
#define MSG_ISSUE(JC, BUF)                                                           \
    {                                                                                \
        _Pragma("unroll")                                                            \
        for (int k = 0; k < 4; ++k) {                                                \
            const int q = tid + 256 * k; /* float4 index */                          \
            __builtin_amdgcn_global_load_async_to_lds_b128(                          \
                (v4i_vs*)(bbase + (size_t)(JC) * H + (size_t)q * 4),                 \
                (__attribute__((address_space(3))) v4i_vs*)&bs_s[BUF][0][q * 4],     \
                0, 0);                                                               \
        }                                                                            \
    }
    MSG_ISSUE(0, 0);
#pragma unroll
    for (int ci = 0; ci < 7; ++ci) {
        const float aw = adj[(size_t)row * NNODE + ci * 64 + hd];
        MSG_ISSUE((ci + 1) * 64, (ci + 1) & 1);   // prefetch next chunk
        adj_s[r][hd] = aw;
        __builtin_amdgcn_s_wait_asynccnt(4);      // chunk ci resident, next in flight
        __syncthreads();
        do_chunk(bs_s[ci & 1]);
        __syncthreads();                          // buffer safe to overwrite at ci+2
    }
    {   // final chunk (ci = 7)
        const float aw = adj[(size_t)row * NNODE + 7 * 64 + hd];
        adj_s[r][hd] = aw;
        __builtin_amdgcn_s_wait_asynccnt(0);
        __syncthreads();
        do_chunk(bs_s[1]);
    }
#else
    for (int ci = 0; ci < 8; ++ci) {
        const int jc = ci * 64;
        const float aw = adj[(size_t)row * NNODE + jc + hd];
        // synchronous fallback: cooperative load of 64 b-rows (16 floats/thread)
#pragma unroll
        for (int k = 0; k < 16; ++k) {
            const int idx = tid + 256 * k;
            bs_s[0][idx >> 6][idx & 63] =
                bbase[(size_t)(jc + (idx >> 6)) * H + (idx & 63)];
        }
        adj_s[r][hd] = aw;
        __syncthreads();
        do_chunk(bs_s[0]);
        __syncthreads();
    }
#endif

    const size_t o = (size_t)row * H + hd;
    h[o] = h[o] + 0.5f * acc / deg[row];
}

// ---------------------------------------------------------------------------
// Scorer: out[row] = mask ? silu(h@ws1+bs1)@ws2 + bs2 : -inf   (one thread/row)
// ---------------------------------------------------------------------------
__global__ __launch_bounds__(256) void score_kernel(
    const float* __restrict__ h, const float* __restrict__ ws1,
    const float* __restrict__ bs1, const float* __restrict__ ws2,
    const float* __restrict__ bs2, const unsigned char* __restrict__ mask,
    float* __restrict__ out)
{
    const int row = blockIdx.x * blockDim.x + threadIdx.x;
    if (row >= NROWS) return;

    float hr[H];
#pragma unroll
    for (int k = 0; k < H; ++k) hr[k] = h[(size_t)row * H + k];

    float score = bs2[0];
#pragma unroll 2
    for (int n = 0; n < 32; ++n) {
        float t = bs1[n];
#pragma unroll
        for (int k = 0; k < H; ++k) t += hr[k] * ws1[k * 32 + n];
        score += silu_f(t) * ws2[n];
    }
    out[row] = mask[row] ? score : -__builtin_inff();
}

// ---------------------------------------------------------------------------
extern "C" void kernel_launch(void* const* d_in, const int* in_sizes, int n_in,
                              void* d_out, int out_size, void* d_ws, size_t ws_size,
                              hipStream_t stream)
{
    const float* features  = (const float*)d_in[0];
    const float* adjacency = (const float*)d_in[1];
    const unsigned char* mask = (const unsigned char*)d_in[2];   // bool array
    const float* fw    = (const float*)d_in[3];
    const float* w_enc = (const float*)d_in[4];
    const float* b_enc = (const float*)d_in[5];
    const float* ln_g  = (const float*)d_in[6];
    const float* ln_b  = (const float*)d_in[7];
    const float* msg_w = (const float*)d_in[8];    // (3, 128, 64)
    const float* msg_b = (const float*)d_in[9];    // (3, 64)
    const float* ws1   = (const float*)d_in[10];
    const float* bs1   = (const float*)d_in[11];
    const float* ws2   = (const float*)d_in[12];
    const float* bs2   = (const float*)d_in[13];
    float* out = (float*)d_out;

    float* h    = (float*)d_ws;            // 4096*64
    float* abuf = h    + (size_t)NROWS * H;
    float* bbuf = abuf + (size_t)NROWS * H;
    float* deg  = bbuf + (size_t)NROWS * H;  // 4096

    enc_kernel<<<NROWS / 16, 32, 0, stream>>>(features, fw, w_enc, b_enc, ln_g, ln_b, h);
    deg_kernel<<<NROWS / 8, 256, 0, stream>>>(adjacency, deg);

    for (int l = 0; l < 3; ++l) {
        const float* wi = msg_w + (size_t)l * 2 * H * H;
        const float* wj = wi + (size_t)H * H;
        gemm64_kernel<<<NROWS / 16, 32, 0, stream>>>(h, wi, nullptr, 0, abuf);
        gemm64_kernel<<<NROWS / 16, 32, 0, stream>>>(h, wj, msg_b + (size_t)l * H, 1, bbuf);
        msg_kernel<<<NROWS / MSG_ROWS, 256, 0, stream>>>(h, abuf, bbuf, adjacency, deg);
    }

    score_kernel<<<NROWS / 256, 256, 0, stream>>>(h, ws1, bs1, ws2, bs2, mask, out);
}